// GCNLayer_2559800508848
// MI455X (gfx1250) — compile-verified
//
#include <hip/hip_runtime.h>
#include <hip/hip_bf16.h>

// GCN layer for MI455X (gfx1250, wave32, WMMA).
//   H = X @ W                      -> fp32 WMMA 16x16x4 GEMM
//   CSR-by-dst build (no fp atomics in hot path)
//   out = dinv_i * sum(dinv_s * H_s) + dinv_i^2 * H_i + b, LeakyReLU

#define DIMD 512
#define NEG_SLOPE 0.01f

typedef __attribute__((ext_vector_type(2))) float v2f;
typedef __attribute__((ext_vector_type(8))) float v8f;

// ---------------------------------------------------------------- CSR build
__global__ void k_zero_count(int* __restrict__ count, int n) {
    int i = blockIdx.x * blockDim.x + threadIdx.x;
    if (i < n) count[i] = 0;
}

__global__ void k_count_edges(const int* __restrict__ dst, int E, int* __restrict__ count) {
    int e = blockIdx.x * blockDim.x + threadIdx.x;
    if (e < E) atomicAdd(&count[dst[e]], 1);
}

// Single-block scan over N (~50K): exclusive prefix -> rowstart/cursor, plus
// dinv[i] = rsqrt(deg_i) with deg_i = in-edges + 1 (self loop).
__global__ void k_scan(const int* __restrict__ count, int* __restrict__ rowstart,
                       int* __restrict__ cursor, float* __restrict__ dinv, int n) {
    __shared__ int sdata[256];
    __shared__ int carry_s;
    if (threadIdx.x == 0) { carry_s = 0; rowstart[0] = 0; }
    __syncthreads();
    for (int base = 0; base < n; base += 256) {
        int i = base + (int)threadIdx.x;
        int v = (i < n) ? count[i] : 0;
        sdata[threadIdx.x] = v;
        __syncthreads();
        for (int off = 1; off < 256; off <<= 1) {
            int t = (threadIdx.x >= (unsigned)off) ? sdata[threadIdx.x - off] : 0;
            __syncthreads();
            sdata[threadIdx.x] += t;
            __syncthreads();
        }
        int incl = sdata[threadIdx.x] + carry_s;   // read old carry
        if (i < n) {
            rowstart[i + 1] = incl;
            cursor[i]       = incl - v;            // exclusive prefix
            dinv[i]         = rsqrtf((float)(v + 1));
        }
        __syncthreads();
        if (threadIdx.x == 0) carry_s += sdata[255];
        __syncthreads();
    }
}

__global__ void k_fill(const int* __restrict__ src, const int* __restrict__ dst, int E,
                       int* __restrict__ cursor, int* __restrict__ srcSorted) {
    int e = blockIdx.x * blockDim.x + threadIdx.x;
    if (e < E) {
        int p = atomicAdd(&cursor[dst[e]], 1);
        srcSorted[p] = src[e];
    }
}

// ---------------------------------------------------------------- GEMM (WMMA)
// Block: 128 threads = 4 waves. Block tile: 64 rows x 64 cols of H.
// Wave w: rows [rowbase + 16w, +16), 4 column tiles of 16. K step = 4.
__global__ __launch_bounds__(128) void k_gemm_wmma(const float* __restrict__ X,
                                                   const float* __restrict__ W,
                                                   float* __restrict__ H, int N) {
    const int lane = threadIdx.x & 31;
    const int wave = threadIdx.x >> 5;
    const int m    = lane & 15;
    const int h2   = (lane >> 4) << 1;              // 0 or 2
    const int rowbase = blockIdx.x * 64 + wave * 16;
    const int colbase = blockIdx.y * 64;

    int arow = rowbase + m;
    if (arow >= N) arow = N - 1;                    // clamp: keep EXEC all-ones
    const float* __restrict__ Arow = X + (size_t)arow * DIMD;

    const int ncol = colbase + (lane & 15);

    v8f acc0 = {}; v8f acc1 = {}; v8f acc2 = {}; v8f acc3 = {};

    #pragma unroll 4
    for (int k0 = 0; k0 < DIMD; k0 += 4) {
        const int k = k0 + h2;
        v2f a;                                       // A 16x4: lane m holds K=2h,2h+1
        a.x = Arow[k];
        a.y = Arow[k + 1];
        const float* __restrict__ Wk0 = W + (size_t)k * DIMD;
        const float* __restrict__ Wk1 = Wk0 + DIMD;
        v2f b0, b1, b2, b3;                          // B 4x16: VGPR r -> K = r + 2h
        b0.x = Wk0[ncol];      b0.y = Wk1[ncol];
        b1.x = Wk0[ncol + 16]; b1.y = Wk1[ncol + 16];
        b2.x = Wk0[ncol + 32]; b2.y = Wk1[ncol + 32];
        b3.x = Wk0[ncol + 48]; b3.y = Wk1[ncol + 48];
        acc0 = __builtin_amdgcn_wmma_f32_16x16x4_f32(false, a, false, b0, (short)0, acc0, false, false);
        acc1 = __builtin_amdgcn_wmma_f32_16x16x4_f32(false, a, false, b1, (short)0, acc1, false, false);
        acc2 = __builtin_amdgcn_wmma_f32_16x16x4_f32(false, a, false, b2, (short)0, acc2, false, false);
        acc3 = __builtin_amdgcn_wmma_f32_16x16x4_f32(false, a, false, b3, (short)0, acc3, false, false);
    }

    // C/D 16x16 f32: VGPR v -> row rowbase + v (+8 for lanes 16..31), col = lane&15.
    const int crow0 = rowbase + ((lane >> 4) << 3);
    const int ccol  = colbase + (lane & 15);
    #pragma unroll
    for (int v = 0; v < 8; v++) {
        int row = crow0 + v;
        if (row < N) {
            float* __restrict__ Hr = H + (size_t)row * DIMD + ccol;
            Hr[0]  = acc0[v];
            Hr[16] = acc1[v];
            Hr[32] = acc2[v];
            Hr[48] = acc3[v];
        }
    }
}

// ---------------------------------------------------------------- aggregate
// One 128-thread block per node; thread t owns float4 at column 4t.
__global__ __launch_bounds__(128) void k_aggregate(const float* __restrict__ H,
                                                   const float* __restrict__ dinv,
                                                   const int* __restrict__ rowstart,
                                                   const int* __restrict__ srcSorted,
                                                   const float* __restrict__ bias,
                                                   float* __restrict__ out, int N) {
    const int node = blockIdx.x;
    if (node >= N) return;
    const int t = threadIdx.x;                       // 0..127
    const float4* __restrict__ H4 = (const float4*)H;

    float ax = 0.f, ay = 0.f, az = 0.f, aw = 0.f;
    const int s = rowstart[node];
    const int e = rowstart[node + 1];
    for (int j = s; j < e; j++) {
        const int src = srcSorted[j];
        const float w = dinv[src];
        const float4 hv = H4[(size_t)src * (DIMD / 4) + t];
        ax += w * hv.x; ay += w * hv.y; az += w * hv.z; aw += w * hv.w;
    }
    const float di  = dinv[node];
    const float di2 = di * di;
    const float4 hs = H4[(size_t)node * (DIMD / 4) + t];
    const float4 bb = ((const float4*)bias)[t];

    float ox = di * ax + di2 * hs.x + bb.x;
    float oy = di * ay + di2 * hs.y + bb.y;
    float oz = di * az + di2 * hs.z + bb.z;
    float ow = di * aw + di2 * hs.w + bb.w;
    ox = (ox >= 0.f) ? ox : NEG_SLOPE * ox;
    oy = (oy >= 0.f) ? oy : NEG_SLOPE * oy;
    oz = (oz >= 0.f) ? oz : NEG_SLOPE * oz;
    ow = (ow >= 0.f) ? ow : NEG_SLOPE * ow;

    float4 o; o.x = ox; o.y = oy; o.z = oz; o.w = ow;
    ((float4*)out)[(size_t)node * (DIMD / 4) + t] = o;
}

// ---------------------------------------------------------------- launch
extern "C" void kernel_launch(void* const* d_in, const int* in_sizes, int n_in,
                              void* d_out, int out_size, void* d_ws, size_t ws_size,
                              hipStream_t stream) {
    const float* X    = (const float*)d_in[0];      // [N, 512]
    const float* W    = (const float*)d_in[1];      // [512, 512]
    const float* bias = (const float*)d_in[2];      // [512]
    const int*   ei   = (const int*)d_in[3];        // [2, E] flat: src then dst
    float*       out  = (float*)d_out;

    const int N = in_sizes[0] / DIMD;
    const int E = in_sizes[3] / 2;
    const int* src = ei;
    const int* dst = ei + E;

    // workspace layout
    char* ws = (char*)d_ws;
    size_t off = 0;
    float* H        = (float*)(ws + off); off += (size_t)N * DIMD * sizeof(float);
    float* dinv     = (float*)(ws + off); off += (size_t)N * sizeof(float);
    int*   count    = (int*)  (ws + off); off += (size_t)N * sizeof(int);
    int*   rowstart = (int*)  (ws + off); off += (size_t)(N + 1) * sizeof(int);
    int*   cursor   = (int*)  (ws + off); off += (size_t)N * sizeof(int);
    int*   srcSort  = (int*)  (ws + off); off += (size_t)E * sizeof(int);
    (void)ws_size; (void)n_in; (void)out_size;

    const int nb_n = (N + 255) / 256;
    const int nb_e = (E + 255) / 256;

    // 1) degree histogram + CSR build
    k_zero_count<<<nb_n, 256, 0, stream>>>(count, N);
    k_count_edges<<<nb_e, 256, 0, stream>>>(dst, E, count);
    k_scan<<<1, 256, 0, stream>>>(count, rowstart, cursor, dinv, N);
    k_fill<<<nb_e, 256, 0, stream>>>(src, dst, E, cursor, srcSort);

    // 2) H = X @ W  (fp32 WMMA)
    dim3 ggrid((N + 63) / 64, DIMD / 64);
    k_gemm_wmma<<<ggrid, 128, 0, stream>>>(X, W, H, N);

    // 3) per-node aggregation + bias + LeakyReLU
    k_aggregate<<<N, 128, 0, stream>>>(H, dinv, rowstart, srcSort, bias, out, N);
}